// HistInstructionEncoder_62130996904131
// MI455X (gfx1250) — compile-verified
//
#include <hip/hip_runtime.h>

// ---------------- problem constants (match reference) ----------------
#define B_SZ   256
#define A_SZ   64
#define FEATN  256
#define H_SZ   16
#define S_SZ   20
#define POSN   64
#define ENCN   256
#define OUTN   384           // 2*POS + ENC
#define MAXD   10

typedef __attribute__((ext_vector_type(2))) float v2f;
typedef __attribute__((ext_vector_type(8))) float v8f;

// ---------------- LDS layout (floats), padded to avoid bank conflicts ----
#define LDA_ARMY 260                       // 64 rows  (260 % 64 == 4)
#define LDP      388                       // 64 rows  (388 % 64 == 4)
#define LDH      388                       // 16 rows
#define LDATT    20                        // 64 rows  (gcd trick, keeps 8B align)
#define OFF_ARMY 0
#define OFF_PARM (OFF_ARMY + A_SZ * LDA_ARMY)   // 16640
#define OFF_HIST (OFF_PARM + A_SZ * LDP)        // 41472
#define OFF_ATT  (OFF_HIST + H_SZ * LDH)        // 47680
#define SMEM_FLOATS (OFF_ATT + A_SZ * LDATT)    // 48960 floats = 195840 bytes

__device__ __forceinline__ v8f wmma4(v2f a, v2f b, v8f c) {
    // V_WMMA_F32_16X16X4_F32: D = A(16x4) * B(4x16) + C(16x16), fp32 throughout
    return __builtin_amdgcn_wmma_f32_16x16x4_f32(
        /*neg_a=*/false, a, /*neg_b=*/false, b,
        /*c_mod=*/(short)0, c, /*reuse_a=*/false, /*reuse_b=*/false);
}

// ---------------- kernel 0: scale = proj_g / ||proj_v|| ----------------
__global__ void norm_scale_kernel(const float* __restrict__ proj_v,
                                  const float* __restrict__ proj_g,
                                  float* __restrict__ scale_out) {
    __shared__ float red[256];
    float s = 0.f;
    for (int i = threadIdx.x; i < OUTN * FEATN; i += 256) {
        float v = proj_v[i];
        s += v * v;
    }
    red[threadIdx.x] = s;
    __syncthreads();
    for (int st = 128; st > 0; st >>= 1) {
        if ((int)threadIdx.x < st) red[threadIdx.x] += red[threadIdx.x + st];
        __syncthreads();
    }
    if (threadIdx.x == 0) scale_out[0] = proj_g[0] * rsqrtf(red[0]);
}

// ---------------- kernel 1: fully fused per-batch attention ----------------
__global__ __launch_bounds__(256) void fused_attn(
    const float* __restrict__ army,     // (B, A, FEAT)
    const int*   __restrict__ inst,     // (B, H, S)
    const int*   __restrict__ ilen,     // (B, H)
    const int*   __restrict__ idiff,    // (B, H)
    const float* __restrict__ tok_emb,  // (VOCAB, ENC)
    const float* __restrict__ pos_w,    // (H, POS)
    const float* __restrict__ diff_w,   // (MAXD+1, POS)
    const float* __restrict__ proj_v,   // (OUT, FEAT)
    const float* __restrict__ proj_b,   // (OUT)
    const float* __restrict__ scale_p,  // (1)  = g/||v||
    float* __restrict__ out_inst,       // (B, A, OUT)
    float* __restrict__ out_sum)        // (B, OUT)
{
    extern __shared__ float smem[];
    float* s_army = smem + OFF_ARMY;   // 64 x FEAT   (padded)
    float* s_parm = smem + OFF_PARM;   // 64 x OUT    (padded)
    float* s_hist = smem + OFF_HIST;   // 16 x OUT    (padded)
    float* s_att  = smem + OFF_ATT;    // 64 x 16     (padded)

    const int b    = blockIdx.x;
    const int tid  = threadIdx.x;
    const int wave = tid >> 5;         // 8 waves (wave32)
    const int lane = tid & 31;
    const int lm   = lane & 15;        // n / m index within frag
    const int lh   = lane >> 4;        // k-half selector
    const float scale = scale_p[0];

    // ---- Phase 0a: build hist_feat[b] = [mean tok_emb | pos | diff] in LDS ----
    for (int e = tid; e < H_SZ * OUTN; e += 256) {
        int h = e / OUTN, d = e - h * OUTN;
        float v;
        if (d < ENCN) {
            int len = ilen[b * H_SZ + h];
            if (len < 1) len = 1;
            const int* row = inst + (b * H_SZ + h) * S_SZ;
            float acc = 0.f;
            for (int s = 0; s < len; ++s) acc += tok_emb[row[s] * ENCN + d];
            v = acc / (float)len;
        } else if (d < ENCN + POSN) {
            v = pos_w[h * POSN + (d - ENCN)];
        } else {
            int di = idiff[b * H_SZ + h];
            di = di < 0 ? 0 : (di > MAXD ? MAXD : di);
            v = diff_w[di * POSN + (d - ENCN - POSN)];
        }
        s_hist[h * LDH + d] = v;
    }
    // ---- Phase 0b: stage army[b] (64 x 256) into LDS ----
    for (int e = tid; e < A_SZ * FEATN; e += 256) {
        int r = e >> 8, c = e & 255;
        s_army[r * LDA_ARMY + c] = army[(size_t)b * A_SZ * FEATN + e];
    }
    __syncthreads();

    // ---- sum_inst output: last history row ----
    for (int d = tid; d < OUTN; d += 256)
        out_sum[b * OUTN + d] = s_hist[(H_SZ - 1) * LDH + d];

    // ---- Phase 1: parm = relu(scale * (army @ proj_v^T) + bias)  [64x384] ----
    // wave w owns n-tiles {3w, 3w+1, 3w+2}, all 4 m-tiles.
    for (int nt = 0; nt < 3; ++nt) {
        const int n0 = (wave * 3 + nt) * 16;
        v8f acc0 = {}, acc1 = {}, acc2 = {}, acc3 = {};
        const float* wrow = proj_v + (size_t)(n0 + lm) * FEATN + 2 * lh;
        for (int k0 = 0; k0 < FEATN; k0 += 4) {
            v2f bf = *(const v2f*)(wrow + k0);
            v2f a0 = *(const v2f*)(&s_army[(0 * 16 + lm) * LDA_ARMY + k0 + 2 * lh]);
            v2f a1 = *(const v2f*)(&s_army[(1 * 16 + lm) * LDA_ARMY + k0 + 2 * lh]);
            v2f a2 = *(const v2f*)(&s_army[(2 * 16 + lm) * LDA_ARMY + k0 + 2 * lh]);
            v2f a3 = *(const v2f*)(&s_army[(3 * 16 + lm) * LDA_ARMY + k0 + 2 * lh]);
            acc0 = wmma4(a0, bf, acc0);
            acc1 = wmma4(a1, bf, acc1);
            acc2 = wmma4(a2, bf, acc2);
            acc3 = wmma4(a3, bf, acc3);
        }
        const float bias = proj_b[n0 + lm];
        #pragma unroll
        for (int r = 0; r < 8; ++r) {
            int mrow = 8 * lh + r;
            float v0 = acc0[r] * scale + bias;
            float v1 = acc1[r] * scale + bias;
            float v2 = acc2[r] * scale + bias;
            float v3 = acc3[r] * scale + bias;
            s_parm[(0 * 16 + mrow) * LDP + n0 + lm] = v0 > 0.f ? v0 : 0.f;
            s_parm[(1 * 16 + mrow) * LDP + n0 + lm] = v1 > 0.f ? v1 : 0.f;
            s_parm[(2 * 16 + mrow) * LDP + n0 + lm] = v2 > 0.f ? v2 : 0.f;
            s_parm[(3 * 16 + mrow) * LDP + n0 + lm] = v3 > 0.f ? v3 : 0.f;
        }
    }
    __syncthreads();

    // ---- Phase 2: dot[a,h] = parm @ hist^T   [64x16], K=384 ----
    if (wave < 4) {                     // wave-uniform branch: EXEC stays all-1s
        const int m0 = wave * 16;
        v8f acc = {};
        for (int k0 = 0; k0 < OUTN; k0 += 4) {
            v2f a  = *(const v2f*)(&s_parm[(m0 + lm) * LDP + k0 + 2 * lh]);
            v2f bf = *(const v2f*)(&s_hist[lm * LDH + k0 + 2 * lh]);
            acc = wmma4(a, bf, acc);
        }
        #pragma unroll
        for (int r = 0; r < 8; ++r)
            s_att[(m0 + 8 * lh + r) * LDATT + lm] = acc[r];
    }
    __syncthreads();

    // ---- softmax over h (16) for each of 64 rows ----
    if (tid < 64) {
        float* row = s_att + tid * LDATT;
        float mx = row[0];
        #pragma unroll
        for (int h = 1; h < 16; ++h) mx = fmaxf(mx, row[h]);
        float sum = 0.f;
        #pragma unroll
        for (int h = 0; h < 16; ++h) { float e = __expf(row[h] - mx); row[h] = e; sum += e; }
        float inv = 1.f / sum;
        #pragma unroll
        for (int h = 0; h < 16; ++h) row[h] *= inv;
    }
    __syncthreads();

    // ---- Phase 3: inst_feat = att @ hist   [64x384], K=16 ----
    for (int nt = 0; nt < 3; ++nt) {
        const int n0 = (wave * 3 + nt) * 16;
        v8f acc0 = {}, acc1 = {}, acc2 = {}, acc3 = {};
        #pragma unroll
        for (int k0 = 0; k0 < 16; k0 += 4) {
            int kk = k0 + 2 * lh;
            v2f bf;
            bf.x = s_hist[kk * LDH + n0 + lm];
            bf.y = s_hist[(kk + 1) * LDH + n0 + lm];
            v2f a0 = *(const v2f*)(&s_att[(0 * 16 + lm) * LDATT + kk]);
            v2f a1 = *(const v2f*)(&s_att[(1 * 16 + lm) * LDATT + kk]);
            v2f a2 = *(const v2f*)(&s_att[(2 * 16 + lm) * LDATT + kk]);
            v2f a3 = *(const v2f*)(&s_att[(3 * 16 + lm) * LDATT + kk]);
            acc0 = wmma4(a0, bf, acc0);
            acc1 = wmma4(a1, bf, acc1);
            acc2 = wmma4(a2, bf, acc2);
            acc3 = wmma4(a3, bf, acc3);
        }
        #pragma unroll
        for (int r = 0; r < 8; ++r) {
            int mrow = 8 * lh + r;
            out_inst[((size_t)b * A_SZ + 0 * 16 + mrow) * OUTN + n0 + lm] = acc0[r];
            out_inst[((size_t)b * A_SZ + 1 * 16 + mrow) * OUTN + n0 + lm] = acc1[r];
            out_inst[((size_t)b * A_SZ + 2 * 16 + mrow) * OUTN + n0 + lm] = acc2[r];
            out_inst[((size_t)b * A_SZ + 3 * 16 + mrow) * OUTN + n0 + lm] = acc3[r];
        }
    }
}

// ---------------- host entry ----------------
extern "C" void kernel_launch(void* const* d_in, const int* in_sizes, int n_in,
                              void* d_out, int out_size, void* d_ws, size_t ws_size,
                              hipStream_t stream) {
    (void)in_sizes; (void)n_in; (void)out_size; (void)ws_size;
    const float* army  = (const float*)d_in[0];
    const int*   inst  = (const int*)  d_in[1];
    const int*   ilen  = (const int*)  d_in[2];
    const int*   idiff = (const int*)  d_in[3];
    const float* tok   = (const float*)d_in[4];
    const float* posw  = (const float*)d_in[5];
    const float* diffw = (const float*)d_in[6];
    const float* pv    = (const float*)d_in[7];
    const float* pg    = (const float*)d_in[8];
    const float* pb    = (const float*)d_in[9];

    float* out      = (float*)d_out;
    float* out_inst = out;                                    // B*A*OUT
    float* out_sum  = out + (size_t)B_SZ * A_SZ * OUTN;       // B*OUT
    float* scale    = (float*)d_ws;                           // 1 float scratch

    // large dynamic LDS (195,840 B) — raise the per-kernel cap (idempotent)
    size_t smem_bytes = (size_t)SMEM_FLOATS * sizeof(float);
    (void)hipFuncSetAttribute((const void*)fused_attn,
                              hipFuncAttributeMaxDynamicSharedMemorySize,
                              (int)smem_bytes);

    hipLaunchKernelGGL(norm_scale_kernel, dim3(1), dim3(256), 0, stream, pv, pg, scale);
    hipLaunchKernelGGL(fused_attn, dim3(B_SZ), dim3(256), smem_bytes, stream,
                       army, inst, ilen, idiff, tok, posw, diffw, pv, pb, scale,
                       out_inst, out_sum);
}